// SINDyLibraryOrder2_55009941127955
// MI455X (gfx1250) — compile-verified
//
#include <hip/hip_runtime.h>
#include <hip/hip_bf16.h>

// Feature layout per row (985 floats):
//   col 0         : 1.0
//   cols 1..16    : zc[0..15]            (zc = [z | dz])
//   cols 17..152  : z_i*z_j, i<=j        (136)
//   cols 153..968 : z_i*z_j*z_k, i<=j<=k (816)
//   cols 969..984 : sin(zc[0..15])
#define NFEAT    985
#define COL_QUAD 17
#define COL_CUBE 153
#define COL_SIN  969
#define NCUBE    816

typedef __attribute__((ext_vector_type(2))) float v2f;
typedef __attribute__((ext_vector_type(8))) float v8f;

// Compile-time tables, passed to the kernel by device pointer (kernarg SGPR
// base -> clean global_load with immediate offsets, no PC-rel materialization,
// no flat/DScnt pollution).
struct alignas(16) Tables {
    // Packed (i,j,k) for the 816 cubic monomials, in
    // itertools.combinations_with_replacement(range(16), 3) order.
    unsigned int idx3[NCUBE];
    // Per-(n, m) canonical quadratic column (u8, max 152), transposed layout
    // qcolT[n*16 + m]: a lane (fixed n, m = mBase..mBase+7) reads 8
    // consecutive bytes -> single aligned b64 load.
    unsigned char qcolT[256];

    constexpr Tables() : idx3(), qcolT() {
        int t = 0;
        for (int a = 0; a < 16; ++a)
            for (int b = a; b < 16; ++b)
                for (int c = b; c < 16; ++c)
                    idx3[t++] = (unsigned)a | ((unsigned)b << 8) | ((unsigned)c << 16);
        for (int m = 0; m < 16; ++m)
            for (int n = 0; n < 16; ++n) {
                const int lo = m < n ? m : n;
                const int hi = m < n ? n : m;
                qcolT[n * 16 + m] =
                    (unsigned char)(COL_QUAD + lo * 16 - ((lo * (lo - 1)) >> 1) + (hi - lo));
            }
    }
};
__device__ Tables g_tab{};

__global__ __launch_bounds__(256) void sindy_lib_kernel(const float* __restrict__ z,
                                                        const float* __restrict__ dz,
                                                        float* __restrict__ out,
                                                        const Tables* __restrict__ tab,
                                                        int nrows) {
    const int lane = threadIdx.x & 31;
    const int wave = threadIdx.x >> 5;
    const int row  = blockIdx.x * 8 + wave;   // wave-uniform
    if (row >= nrows) return;                 // whole wave exits together -> EXEC stays full

    float* __restrict__ rowp = out + (size_t)row * NFEAT;

    // Branch-free input load: lanes 0..7 -> z, 8..15 -> dz (lanes 16..31 load
    // a duplicate and are zeroed; they are never consumed).
    const float* src = (lane & 8) ? dz : z;
    float vz = src[(size_t)row * 8 + (lane & 7)];
    vz = (lane < 16) ? vz : 0.0f;

    // ---- constant + linear + sine columns ----
    if (lane < 16) {
        rowp[1 + lane]       = vz;
        rowp[COL_SIN + lane] = __sinf(vz);
    } else if (lane == 16) {
        rowp[0] = 1.0f;
    }

    // ---- quadratic block: rank-1 outer product on the matrix pipe ----
    // A (16x4 f32): VGPR0 lanes0-15 = K=0 column = zc[M]; all else 0.
    // B (4x16 f32): VGPR0 lanes0-15 = K=0 row    = zc[N]; all else 0.
    // => D[m][n] = zc[m] * zc[n]  (only K=0 contributes).
    v2f a;
    a.x = vz;          // already 0 for lanes >= 16
    a.y = 0.0f;
    v2f b = a;
    v8f c = {};
    c = __builtin_amdgcn_wmma_f32_16x16x4_f32(
            /*neg_a=*/false, a, /*neg_b=*/false, b,
            /*c_mod=*/(short)0, c, /*reuse_a=*/false, /*reuse_b=*/false);

    // C/D layout: VGPR v -> M = mBase+v (mBase = 0 / 8), N = lane&15.
    // Columns come from the u8 LUT; both triangle halves map to the same slot
    // and hold the identical product, so duplicate stores are benign.
    {
        const unsigned char* qt =
            tab->qcolT + ((lane & 15) << 4) + ((lane >> 4) << 3);  // 8B aligned
        const uint2 q = *(const uint2*)qt;
#pragma unroll
        for (int v = 0; v < 8; ++v) {
            const unsigned col =
                ((v < 4 ? q.x >> (8 * v) : q.y >> (8 * (v - 4))) & 0xFFu);
            rowp[col] = c[v];
        }
    }

    // ---- cubic block: straight-line, immediate-offset coalesced stores ----
    {
        const unsigned* __restrict__ pt = tab->idx3;   // kernarg SGPR base
        float* __restrict__ pc = rowp + COL_CUBE + lane;
#pragma unroll
        for (int k = 0; k < 25; ++k) {                 // t = lane + 32k <= 799+lane
            const unsigned p = pt[lane + 32 * k];
            const float zi = __shfl(vz, (int)(p & 0xFFu), 32);
            const float zj = __shfl(vz, (int)((p >> 8) & 0xFFu), 32);
            const float zk = __shfl(vz, (int)((p >> 16) & 0xFFu), 32);
            pc[32 * k] = zi * zj * zk;
        }
        // tail: t = 800 + lane, valid for lanes 0..15 (816 total)
        const unsigned p = pt[lane + 800];
        const float zi = __shfl(vz, (int)(p & 0xFFu), 32);
        const float zj = __shfl(vz, (int)((p >> 8) & 0xFFu), 32);
        const float zk = __shfl(vz, (int)((p >> 16) & 0xFFu), 32);
        if (lane < 16) pc[800] = zi * zj * zk;
    }
}

extern "C" void kernel_launch(void* const* d_in, const int* in_sizes, int n_in,
                              void* d_out, int out_size, void* d_ws, size_t ws_size,
                              hipStream_t stream) {
    const float* z  = (const float*)d_in[0];
    const float* dz = (const float*)d_in[1];
    float* out      = (float*)d_out;

    void* tabp = nullptr;
    (void)hipGetSymbolAddress(&tabp, HIP_SYMBOL(g_tab));  // pure query; capture-safe

    const int nrows  = in_sizes[0] / 8;          // LATENT_DIM = 8
    const int blocks = (nrows + 7) / 8;          // 8 waves (rows) per 256-thread block
    sindy_lib_kernel<<<blocks, 256, 0, stream>>>(z, dz, out, (const Tables*)tabp, nrows);
}